// IOU3Dloss_48704929136735
// MI455X (gfx1250) — compile-verified
//
#include <hip/hip_runtime.h>
#include <math.h>

#define EPS_HULL 1e-5f

typedef float v2f __attribute__((ext_vector_type(2)));
typedef float v8f __attribute__((ext_vector_type(8)));

// Predicated scatter into a register-resident polygon slot (keeps everything
// out of scratch: constant indices only, selection by compare).
__device__ __forceinline__ void emit_vert(float (&ox)[8], float (&oy)[8], int &oc,
                                          float vx, float vy) {
#pragma unroll
  for (int s = 0; s < 8; ++s)
    if (oc == s) { ox[s] = vx; oy[s] = vy; }
  ++oc;
}

__global__ __launch_bounds__(256) void giou3d_main(const float* __restrict__ pred,
                                                   const float* __restrict__ tgt,
                                                   float* __restrict__ partials,
                                                   int n, float inv_n) {
  const int i = blockIdx.x * blockDim.x + threadIdx.x;
  float local = 0.f;
  if (i < n) {
    const float* P = pred + 7 * i;
    const float* T = tgt + 7 * i;
    __builtin_prefetch(T, 0, 1);  // global_prefetch_b8 on gfx1250

    float px = P[0], py = P[1], pz = P[2], ph = P[3], pw = P[4], pl = P[5], pyaw = P[6];
    float tx = T[0], ty = T[1], tz = T[2], th = T[3], tw = T[4], tl = T[5], tyaw = T[6];

    // --- rotated BEV corners (CCW), matching reference sign convention ---
    const float sx[4] = {1.f, -1.f, -1.f, 1.f};
    const float sy[4] = {1.f, 1.f, -1.f, -1.f};
    float c0 = cosf(pyaw), s0 = sinf(pyaw);
    float c1 = cosf(tyaw), s1 = sinf(tyaw);
    float pcx[4], pcy[4], tcx[4], tcy[4];
#pragma unroll
    for (int k = 0; k < 4; ++k) {
      float dx = 0.5f * pw * sx[k], dy = 0.5f * pl * sy[k];
      pcx[k] = px + dx * c0 - dy * s0;
      pcy[k] = py + dx * s0 + dy * c0;
      float ex = 0.5f * tw * sx[k], ey = 0.5f * tl * sy[k];
      tcx[k] = tx + ex * c1 - ey * s1;
      tcy[k] = ty + ex * s1 + ey * c1;
    }

    // --- Sutherland-Hodgman clip of pred quad by target quad (4 half-planes) ---
    float qx[8], qy[8];
#pragma unroll
    for (int k = 0; k < 8; ++k) { qx[k] = (k < 4) ? pcx[k] : 0.f; qy[k] = (k < 4) ? pcy[k] : 0.f; }
    int cnt = 4;
#pragma unroll
    for (int e = 0; e < 4; ++e) {
      float ax = tcx[e], ay = tcy[e];
      float bx = tcx[(e + 1) & 3], by = tcy[(e + 1) & 3];
      float abx = bx - ax, aby = by - ay;
      float ox[8], oy[8];
#pragma unroll
      for (int k = 0; k < 8; ++k) { ox[k] = 0.f; oy[k] = 0.f; }
      int oc = 0;
#pragma unroll
      for (int j = 0; j < 8; ++j) {
        if (j < cnt) {
          float cx_ = qx[j], cy_ = qy[j];
          // next vertex: j+1 if still valid, else wrap to 0 (j<7 is compile-time)
          bool adv = (j < 7) && (j + 1 < cnt);
          float nx_ = adv ? qx[(j < 7) ? j + 1 : 0] : qx[0];
          float ny_ = adv ? qy[(j < 7) ? j + 1 : 0] : qy[0];
          float s_cur = abx * (cy_ - ay) - aby * (cx_ - ax);
          float s_nxt = abx * (ny_ - ay) - aby * (nx_ - ax);
          bool in_c = (s_cur >= 0.f);
          bool in_n = (s_nxt >= 0.f);
          float denom = s_cur - s_nxt;
          float t = s_cur / ((fabsf(denom) > 1e-12f) ? denom : 1.f);
          float ix = cx_ + t * (nx_ - cx_);
          float iy = cy_ + t * (ny_ - cy_);
          if (in_c) emit_vert(ox, oy, oc, cx_, cy_);
          if (in_c != in_n) emit_vert(ox, oy, oc, ix, iy);
        }
      }
#pragma unroll
      for (int k = 0; k < 8; ++k) { qx[k] = ox[k]; qy[k] = oy[k]; }
      cnt = oc;
    }

    // --- shoelace area of intersection polygon ---
    float acc = 0.f;
#pragma unroll
    for (int j = 0; j < 8; ++j) {
      if (j < cnt) {
        bool adv = (j < 7) && (j + 1 < cnt);
        float nx_ = adv ? qx[(j < 7) ? j + 1 : 0] : qx[0];
        float ny_ = adv ? qy[(j < 7) ? j + 1 : 0] : qy[0];
        acc += qx[j] * ny_ - qy[j] * nx_;
      }
    }
    float inter_area = 0.5f * fabsf(acc);

    // --- vertical overlap, IoU ---
    float p_low = pz - 0.5f * ph, p_high = pz + 0.5f * ph;
    float t_low = tz - 0.5f * th, t_high = tz + 0.5f * th;
    float inter_h = fmaxf(0.f, fminf(p_high, t_high) - fmaxf(p_low, t_low));
    float inter_vol = inter_h * inter_area;
    float uni = ph * pw * pl + th * tw * tl - inter_vol;
    float iou = inter_vol / (uni + 1e-16f);

    // --- convex hull area of 8 points (all-pairs edge test, same as reference) ---
    float hx[8], hy[8];
#pragma unroll
    for (int k = 0; k < 4; ++k) {
      hx[k] = pcx[k]; hy[k] = pcy[k];
      hx[k + 4] = tcx[k]; hy[k + 4] = tcy[k];
    }
    float hs = 0.f;
#pragma unroll
    for (int a = 0; a < 8; ++a) {
#pragma unroll
      for (int b = 0; b < 8; ++b) {
        if (a == b) continue;
        float dxab = hx[b] - hx[a], dyab = hy[b] - hy[a];
        bool edge = true;
#pragma unroll
        for (int k = 0; k < 8; ++k) {
          float dxak = hx[k] - hx[a], dyak = hy[k] - hy[a];
          edge = edge && (dxab * dyak - dyab * dxak >= -EPS_HULL);
        }
        if (edge) hs += hx[a] * hy[b] - hx[b] * hy[a];
      }
    }
    float hull_area = 0.5f * fabsf(hs);

    float convex_h = fmaxf(0.f, fmaxf(p_high, t_high) - fminf(p_low, t_low));
    float convex_vol = convex_h * hull_area;
    float giou = iou - (convex_vol - uni) / (convex_vol + 1e-16f);
    local = (1.f - giou) * inv_n;
  }

  // deterministic per-block tree reduction (graph replays must be bit-stable)
  __shared__ float red[256];
  red[threadIdx.x] = local;
  __syncthreads();
#pragma unroll
  for (int s = 128; s > 0; s >>= 1) {
    if ((int)threadIdx.x < s) red[threadIdx.x] += red[threadIdx.x + s];
    __syncthreads();
  }
  if (threadIdx.x == 0) partials[blockIdx.x] = red[0];
}

// Single-wave final reduction. Uses V_WMMA_F32_16X16X4_F32 with an all-ones
// A matrix as an exact f32 summation engine: with A==1, D[m,n] = sum_k B[k,n],
// so folding one row of D (16 lanes) yields the sum of ALL entries of B
// regardless of the lane->(k,n) mapping. No precision loss (pure f32 adds).
__global__ void giou3d_reduce(const float* __restrict__ partials, int nparts,
                              float* __restrict__ out) {
  const int lane = threadIdx.x;  // exactly one wave32
  float v = 0.f;
  for (int j = lane; j < nparts; j += 32) v += partials[j];

#if __has_builtin(__builtin_amdgcn_wmma_f32_16x16x4_f32)
  v2f a; a[0] = 1.f; a[1] = 1.f;       // A = ones(16x4)
  v2f b; b[0] = v;   b[1] = 0.f;       // 32 per-lane partials live in B
  v8f c = {};
  v8f d = __builtin_amdgcn_wmma_f32_16x16x4_f32(
      /*neg_a=*/false, a, /*neg_b=*/false, b,
      /*c_mod=*/(short)0, c, /*reuse_a=*/false, /*reuse_b=*/false);
  float r = d[0];                       // VGPR0: lanes 0..15 hold the 16 column sums
#pragma unroll
  for (int off = 8; off >= 1; off >>= 1) r += __shfl_xor(r, off, 16);
#else
  float r = v;
#pragma unroll
  for (int off = 16; off >= 1; off >>= 1) r += __shfl_xor(r, off, 32);
#endif
  if (lane == 0) out[0] = r;
}

extern "C" void kernel_launch(void* const* d_in, const int* in_sizes, int n_in,
                              void* d_out, int out_size, void* d_ws, size_t ws_size,
                              hipStream_t stream) {
  const float* pred = (const float*)d_in[0];
  const float* tgt  = (const float*)d_in[1];
  float* out = (float*)d_out;
  float* partials = (float*)d_ws;

  const int n = in_sizes[0] / 7;                 // 131072 boxes
  const int blocks = (n + 255) / 256;            // 512 partial sums -> d_ws

  giou3d_main<<<blocks, 256, 0, stream>>>(pred, tgt, partials, n, 1.f / (float)n);
  giou3d_reduce<<<1, 32, 0, stream>>>(partials, blocks, out);
}